// DynamicFeatureFusion_old_69518340653173
// MI455X (gfx1250) — compile-verified
//
#include <hip/hip_runtime.h>

#define DIM   4096
#define NROWS 8192
#define TWOD  8192

typedef float v2f __attribute__((ext_vector_type(2)));
typedef float v8f __attribute__((ext_vector_type(8)));

#define BM 128
#define BN 128
#define BK 16
// floats per LDS pair-row; 288 mod 64 == 32 -> the two 16-lane halves of a
// wmma fragment b64 load hit disjoint bank ranges (conflict-free).
#define PSTR 288
#define TILEF (8 * PSTR)   // 8 k-pairs per BK=16 tile
// row stride (floats) for the row-major async A tile: 20*dr ≡ 2 (mod 64) has
// no solution -> fragment b64 loads are bank-conflict-free across both halves.
#define ASTR 20

#if defined(__HIP_DEVICE_COMPILE__) && __has_builtin(__builtin_amdgcn_global_load_async_to_lds_b128)
#define HAVE_ASYNC 1
#else
#define HAVE_ASYNC 0
#endif

#if HAVE_ASYNC
typedef int gcc_v4i __attribute__((__vector_size__(4 * sizeof(int))));
__device__ __forceinline__ void async_copy16(const float* g, float* l) {
    __builtin_amdgcn_global_load_async_to_lds_b128(
        (__attribute__((address_space(1))) gcc_v4i*)g,
        (__attribute__((address_space(3))) gcc_v4i*)l, 0, 0);
}
#if __has_builtin(__builtin_amdgcn_s_wait_asynccnt)
#define WAIT_ASYNC(n) __builtin_amdgcn_s_wait_asynccnt(n)
#else
#define WAIT_ASYNC(n) asm volatile("s_wait_asynccnt %0" ::"i"(n) : "memory")
#endif
#endif

// ---------------------------------------------------------------------------
// LayerNorm: one block per row, 256 threads, 16 floats/thread (4x float4)
// ---------------------------------------------------------------------------
__global__ __launch_bounds__(256) void ln_kernel(const float* __restrict__ x,
                                                 const float* __restrict__ g,
                                                 const float* __restrict__ bb,
                                                 float* __restrict__ y) {
    __shared__ float red[16];
    const int tid = threadIdx.x;
    const int row = blockIdx.x;
    const float4* xr = (const float4*)(x + (size_t)row * DIM);
    float4 v[4];
    float s = 0.f, sq = 0.f;
#pragma unroll
    for (int i = 0; i < 4; ++i) {
        v[i] = xr[tid + 256 * i];
        s  += v[i].x + v[i].y + v[i].z + v[i].w;
        sq += v[i].x * v[i].x + v[i].y * v[i].y + v[i].z * v[i].z + v[i].w * v[i].w;
    }
#pragma unroll
    for (int off = 16; off > 0; off >>= 1) {
        s  += __shfl_xor(s, off, 32);
        sq += __shfl_xor(sq, off, 32);
    }
    const int wid = tid >> 5;
    if ((tid & 31) == 0) { red[wid * 2] = s; red[wid * 2 + 1] = sq; }
    __syncthreads();
    float st = 0.f, sqt = 0.f;
#pragma unroll
    for (int w = 0; w < 8; ++w) { st += red[w * 2]; sqt += red[w * 2 + 1]; }
    const float mu   = st * (1.0f / DIM);
    const float var  = sqt * (1.0f / DIM) - mu * mu;
    const float rstd = rsqrtf(var + 1e-5f);
    float4* yr = (float4*)(y + (size_t)row * DIM);
    const float4* gv4 = (const float4*)g;
    const float4* bv4 = (const float4*)bb;
#pragma unroll
    for (int i = 0; i < 4; ++i) {
        const int c4 = tid + 256 * i;
        const float4 gv = gv4[c4];
        const float4 bv = bv4[c4];
        float4 o;
        o.x = (v[i].x - mu) * rstd * gv.x + bv.x;
        o.y = (v[i].y - mu) * rstd * gv.y + bv.y;
        o.z = (v[i].z - mu) * rstd * gv.z + bv.z;
        o.w = (v[i].w - mu) * rstd * gv.w + bv.w;
        yr[c4] = o;
    }
}

// ---------------------------------------------------------------------------
// Shared GEMM helpers.
// Pair-interleaved LDS layout: element (k, n) at buf[(k>>1)*PSTR + 2n + (k&1)]
// -> a wmma fragment (k0, k0+1) at column n is one contiguous float2.
// ---------------------------------------------------------------------------
__device__ __forceinline__ void store_a_tile(float* base, int k4, int r0,
                                             const float4 a0, const float4 a1) {
    const int p = k4 >> 1;  // 0,2,4,6
    *(float2*)&base[(p + 0) * PSTR + 2 * r0]        = make_float2(a0.x, a0.y);
    *(float2*)&base[(p + 1) * PSTR + 2 * r0]        = make_float2(a0.z, a0.w);
    *(float2*)&base[(p + 0) * PSTR + 2 * (r0 + 64)] = make_float2(a1.x, a1.y);
    *(float2*)&base[(p + 1) * PSTR + 2 * (r0 + 64)] = make_float2(a1.z, a1.w);
}

__device__ __forceinline__ void store_b_tile(float* base, int pB, int m4,
                                             const float4 b0, const float4 b1) {
    float* dst = &base[pB * PSTR + 2 * m4];
    float4 s0, s1;
    s0.x = b0.x; s0.y = b1.x; s0.z = b0.y; s0.w = b1.y;
    s1.x = b0.z; s1.y = b1.z; s1.z = b0.w; s1.w = b1.w;
    *(float4*)&dst[0] = s0;
    *(float4*)&dst[4] = s1;
}

// ---------------------------------------------------------------------------
// GEMM1: h = relu([t_norm|k_norm] @ W1 + b1)  -- h NOT stored.
// Epilogue: partial logits l[n,c] += sum_m h[n,m]*W2[m,c], deterministic,
// per column-block to partial[32][N][2].
// Block tile 128x128, 8 waves, wave tile 32x64 (2x4 wmma subtiles).
// A tile: async global->LDS (row-major, ASYNCcnt).  B tile: register-staged
// k-pair interleave (DScnt).  Double-buffered.
// ---------------------------------------------------------------------------
__global__ __launch_bounds__(256) void gemm1_kernel(const float* __restrict__ tn,
                                                    const float* __restrict__ kn,
                                                    const float* __restrict__ W1,
                                                    const float* __restrict__ b1,
                                                    const float* __restrict__ W2,
                                                    float* __restrict__ partial) {
#if HAVE_ASYNC
    __shared__ __align__(16) float Aa[2][BM * ASTR];   // row-major: [row][k]
#else
    __shared__ __align__(16) float As[2][TILEF];
#endif
    __shared__ __align__(16) float Bs[2][TILEF];
    __shared__ float sh[2][BM][2];

    const int tid  = threadIdx.x;
    const int lane = tid & 31;
    const int wid  = tid >> 5;
    const int wm   = wid & 3;    // 4 row-groups of 32
    const int wn   = wid >> 2;   // 2 col-groups of 64
    const int bj   = blockIdx.x; // 0..31
    const int bi   = blockIdx.y; // 0..63
    const int half = lane >> 4, lr = lane & 15;

    const int r0 = tid >> 2;          // 0..63
    const int k4 = (tid & 3) << 2;    // 0,4,8,12
    const int pB = tid >> 5;          // 0..7
    const int m4 = (tid & 31) << 2;   // 0..124

    const size_t aRow0 = (size_t)(bi * BM + r0) * DIM;
    const size_t aRow1 = (size_t)(bi * BM + r0 + 64) * DIM;
    const size_t bCol  = (size_t)bj * BN + m4;

    v8f acc[2][4];
#pragma unroll
    for (int i = 0; i < 2; ++i)
#pragma unroll
        for (int j = 0; j < 4; ++j) acc[i][j] = (v8f)0.0f;

    const int T = TWOD / BK;  // 512

#if HAVE_ASYNC
    float4 bR0, bR1;
    // prologue: async A(0), B regs(0) + store
    async_copy16(tn + aRow0 + k4, &Aa[0][r0 * ASTR + k4]);
    async_copy16(tn + aRow1 + k4, &Aa[0][(r0 + 64) * ASTR + k4]);
    bR0 = *(const float4*)(W1 + (size_t)(2 * pB) * DIM + bCol);
    bR1 = *(const float4*)(W1 + (size_t)(2 * pB + 1) * DIM + bCol);
    store_b_tile(Bs[0], pB, m4, bR0, bR1);

    for (int t = 0; t < T; ++t) {
        const int buf  = t & 1;
        const bool more = (t + 1 < T);
        if (more) {
            const int kt = (t + 1) * BK;
            bR0 = *(const float4*)(W1 + (size_t)(kt + 2 * pB) * DIM + bCol);
            bR1 = *(const float4*)(W1 + (size_t)(kt + 2 * pB + 1) * DIM + bCol);
            if (t + 4 < T)
                __builtin_prefetch(W1 + (size_t)((t + 4) * BK + 2 * pB) * DIM + bCol, 0, 3);
        }
        __syncthreads();   // all waves done computing tile t-1 (frees buf^1)
        if (more) {
            const int kt   = (t + 1) * BK;
            const float* src  = (kt < DIM) ? tn : kn;
            const int    kofs = (kt < DIM) ? kt : kt - DIM;
            async_copy16(src + aRow0 + kofs + k4, &Aa[buf ^ 1][r0 * ASTR + k4]);
            async_copy16(src + aRow1 + kofs + k4, &Aa[buf ^ 1][(r0 + 64) * ASTR + k4]);
            store_b_tile(Bs[buf ^ 1], pB, m4, bR0, bR1);
            WAIT_ASYNC(2);   // tile t's two async copies done (in-order)
        } else {
            WAIT_ASYNC(0);
        }
        __syncthreads();   // tile t fully visible to all waves
#pragma unroll
        for (int kk = 0; kk < BK; kk += 4) {
            const int k0 = kk + 2 * half;
            const float* bb = &Bs[buf][((kk >> 1) + half) * PSTR];
            v2f af[2], bf[4];
#pragma unroll
            for (int i = 0; i < 2; ++i)
                af[i] = *(const v2f*)&Aa[buf][(wm * 32 + i * 16 + lr) * ASTR + k0];
#pragma unroll
            for (int j = 0; j < 4; ++j)
                bf[j] = *(const v2f*)&bb[2 * (wn * 64 + j * 16 + lr)];
#pragma unroll
            for (int i = 0; i < 2; ++i)
#pragma unroll
                for (int j = 0; j < 4; ++j)
                    acc[i][j] = __builtin_amdgcn_wmma_f32_16x16x4_f32(
                        false, af[i], false, bf[j], (short)0, acc[i][j], false, false);
        }
    }
#else
    float4 aR0, aR1, bR0, bR1;
    aR0 = *(const float4*)(tn + aRow0 + k4);
    aR1 = *(const float4*)(tn + aRow1 + k4);
    bR0 = *(const float4*)(W1 + (size_t)(2 * pB) * DIM + bCol);
    bR1 = *(const float4*)(W1 + (size_t)(2 * pB + 1) * DIM + bCol);
    store_a_tile(As[0], k4, r0, aR0, aR1);
    store_b_tile(Bs[0], pB, m4, bR0, bR1);

    for (int t = 0; t < T; ++t) {
        const int buf  = t & 1;
        const bool more = (t + 1 < T);
        if (more) {
            const int kt   = (t + 1) * BK;
            const float* src  = (kt < DIM) ? tn : kn;
            const int    kofs = (kt < DIM) ? kt : kt - DIM;
            aR0 = *(const float4*)(src + aRow0 + kofs + k4);
            aR1 = *(const float4*)(src + aRow1 + kofs + k4);
            bR0 = *(const float4*)(W1 + (size_t)(kt + 2 * pB) * DIM + bCol);
            bR1 = *(const float4*)(W1 + (size_t)(kt + 2 * pB + 1) * DIM + bCol);
            if (t + 4 < T)
                __builtin_prefetch(W1 + (size_t)((t + 4) * BK + 2 * pB) * DIM + bCol, 0, 3);
        }
        __syncthreads();
        if (more) {
            store_a_tile(As[buf ^ 1], k4, r0, aR0, aR1);
            store_b_tile(Bs[buf ^ 1], pB, m4, bR0, bR1);
        }
#pragma unroll
        for (int kk = 0; kk < BK; kk += 4) {
            const int p = (kk >> 1) + half;
            const float* ab = &As[buf][p * PSTR];
            const float* bb = &Bs[buf][p * PSTR];
            v2f af[2], bf[4];
#pragma unroll
            for (int i = 0; i < 2; ++i)
                af[i] = *(const v2f*)&ab[2 * (wm * 32 + i * 16 + lr)];
#pragma unroll
            for (int j = 0; j < 4; ++j)
                bf[j] = *(const v2f*)&bb[2 * (wn * 64 + j * 16 + lr)];
#pragma unroll
            for (int i = 0; i < 2; ++i)
#pragma unroll
                for (int j = 0; j < 4; ++j)
                    acc[i][j] = __builtin_amdgcn_wmma_f32_16x16x4_f32(
                        false, af[i], false, bf[j], (short)0, acc[i][j], false, false);
        }
    }
#endif

    // Epilogue: bias + relu, dot with W2 columns, lane reduce, stage in LDS.
    float b1v[4], w20[4], w21[4];
#pragma unroll
    for (int j = 0; j < 4; ++j) {
        const int gcol = bj * BN + wn * 64 + j * 16 + lr;
        b1v[j] = b1[gcol];
        w20[j] = W2[gcol * 2 + 0];
        w21[j] = W2[gcol * 2 + 1];
    }
    __syncthreads();
#pragma unroll
    for (int i = 0; i < 2; ++i) {
#pragma unroll
        for (int r = 0; r < 8; ++r) {
            float s0 = 0.f, s1 = 0.f;
#pragma unroll
            for (int j = 0; j < 4; ++j) {
                const float h = fmaxf(acc[i][j][r] + b1v[j], 0.0f);
                s0 += h * w20[j];
                s1 += h * w21[j];
            }
#pragma unroll
            for (int off = 1; off < 16; off <<= 1) {
                s0 += __shfl_xor(s0, off, 32);
                s1 += __shfl_xor(s1, off, 32);
            }
            if (lr == 0) {
                const int lrow = wm * 32 + i * 16 + r + 8 * half;
                sh[wn][lrow][0] = s0;
                sh[wn][lrow][1] = s1;
            }
        }
    }
    __syncthreads();
    {
        const int lrow = tid >> 1;
        const int c    = tid & 1;
        const float v  = sh[0][lrow][c] + sh[1][lrow][c];
        partial[((size_t)bj * NROWS + (bi * BM + lrow)) * 2 + c] = v;
    }
}

// ---------------------------------------------------------------------------
// Softmax over [N,2]: deterministic fixed-order sum of the 32 partials.
// ---------------------------------------------------------------------------
__global__ __launch_bounds__(256) void softmax_kernel(const float* __restrict__ partial,
                                                      const float* __restrict__ b2,
                                                      float* __restrict__ wts) {
    const int n = blockIdx.x * 256 + threadIdx.x;
    float l0 = b2[0], l1 = b2[1];
#pragma unroll 4
    for (int cb = 0; cb < 32; ++cb) {
        l0 += partial[((size_t)cb * NROWS + n) * 2 + 0];
        l1 += partial[((size_t)cb * NROWS + n) * 2 + 1];
    }
    const float m  = fmaxf(l0, l1);
    const float e0 = __expf(l0 - m), e1 = __expf(l1 - m);
    const float inv = 1.0f / (e0 + e1);
    wts[n * 2 + 0] = e0 * inv;
    wts[n * 2 + 1] = e1 * inv;
}

// ---------------------------------------------------------------------------
// GEMM out: out = (w0*t_norm + w1*k_norm) @ Wout + bout.
// Fused A computed on the fly while filling LDS (saves a full HBM pass).
// ---------------------------------------------------------------------------
__global__ __launch_bounds__(256) void gemm_out_kernel(const float* __restrict__ tn,
                                                       const float* __restrict__ kn,
                                                       const float* __restrict__ wts,
                                                       const float* __restrict__ Wout,
                                                       const float* __restrict__ bout,
                                                       float* __restrict__ out) {
    __shared__ __align__(16) float As[2][TILEF];
    __shared__ __align__(16) float Bs[2][TILEF];

    const int tid  = threadIdx.x;
    const int lane = tid & 31;
    const int wid  = tid >> 5;
    const int wm   = wid & 3;
    const int wn   = wid >> 2;
    const int bj   = blockIdx.x;
    const int bi   = blockIdx.y;
    const int half = lane >> 4, lr = lane & 15;

    const int r0 = tid >> 2;
    const int k4 = (tid & 3) << 2;
    const int pB = tid >> 5;
    const int m4 = (tid & 31) << 2;

    const size_t aRow0 = (size_t)(bi * BM + r0) * DIM;
    const size_t aRow1 = (size_t)(bi * BM + r0 + 64) * DIM;
    const size_t bCol  = (size_t)bj * BN + m4;

    const float w0a = wts[(bi * BM + r0) * 2 + 0];
    const float w1a = wts[(bi * BM + r0) * 2 + 1];
    const float w0b = wts[(bi * BM + r0 + 64) * 2 + 0];
    const float w1b = wts[(bi * BM + r0 + 64) * 2 + 1];

    v8f acc[2][4];
#pragma unroll
    for (int i = 0; i < 2; ++i)
#pragma unroll
        for (int j = 0; j < 4; ++j) acc[i][j] = (v8f)0.0f;

    float4 aR0, aR1, bR0, bR1;
    {
        const float4 a0 = *(const float4*)(tn + aRow0 + k4);
        const float4 c0 = *(const float4*)(kn + aRow0 + k4);
        const float4 a1 = *(const float4*)(tn + aRow1 + k4);
        const float4 c1 = *(const float4*)(kn + aRow1 + k4);
        aR0.x = w0a * a0.x + w1a * c0.x; aR0.y = w0a * a0.y + w1a * c0.y;
        aR0.z = w0a * a0.z + w1a * c0.z; aR0.w = w0a * a0.w + w1a * c0.w;
        aR1.x = w0b * a1.x + w1b * c1.x; aR1.y = w0b * a1.y + w1b * c1.y;
        aR1.z = w0b * a1.z + w1b * c1.z; aR1.w = w0b * a1.w + w1b * c1.w;
        bR0 = *(const float4*)(Wout + (size_t)(2 * pB) * DIM + bCol);
        bR1 = *(const float4*)(Wout + (size_t)(2 * pB + 1) * DIM + bCol);
    }
    store_a_tile(As[0], k4, r0, aR0, aR1);
    store_b_tile(Bs[0], pB, m4, bR0, bR1);

    const int T = DIM / BK;  // 256
    for (int t = 0; t < T; ++t) {
        const int buf  = t & 1;
        const bool more = (t + 1 < T);
        if (more) {
            const int kt = (t + 1) * BK;
            const float4 a0 = *(const float4*)(tn + aRow0 + kt + k4);
            const float4 c0 = *(const float4*)(kn + aRow0 + kt + k4);
            const float4 a1 = *(const float4*)(tn + aRow1 + kt + k4);
            const float4 c1 = *(const float4*)(kn + aRow1 + kt + k4);
            aR0.x = w0a * a0.x + w1a * c0.x; aR0.y = w0a * a0.y + w1a * c0.y;
            aR0.z = w0a * a0.z + w1a * c0.z; aR0.w = w0a * a0.w + w1a * c0.w;
            aR1.x = w0b * a1.x + w1b * c1.x; aR1.y = w0b * a1.y + w1b * c1.y;
            aR1.z = w0b * a1.z + w1b * c1.z; aR1.w = w0b * a1.w + w1b * c1.w;
            bR0 = *(const float4*)(Wout + (size_t)(kt + 2 * pB) * DIM + bCol);
            bR1 = *(const float4*)(Wout + (size_t)(kt + 2 * pB + 1) * DIM + bCol);
            if (t + 4 < T)
                __builtin_prefetch(Wout + (size_t)((t + 4) * BK + 2 * pB) * DIM + bCol, 0, 3);
        }
        __syncthreads();
        if (more) {
            store_a_tile(As[buf ^ 1], k4, r0, aR0, aR1);
            store_b_tile(Bs[buf ^ 1], pB, m4, bR0, bR1);
        }
#pragma unroll
        for (int kk = 0; kk < BK; kk += 4) {
            const int p = (kk >> 1) + half;
            const float* ab = &As[buf][p * PSTR];
            const float* bb = &Bs[buf][p * PSTR];
            v2f af[2], bf[4];
#pragma unroll
            for (int i = 0; i < 2; ++i)
                af[i] = *(const v2f*)&ab[2 * (wm * 32 + i * 16 + lr)];
#pragma unroll
            for (int j = 0; j < 4; ++j)
                bf[j] = *(const v2f*)&bb[2 * (wn * 64 + j * 16 + lr)];
#pragma unroll
            for (int i = 0; i < 2; ++i)
#pragma unroll
                for (int j = 0; j < 4; ++j)
                    acc[i][j] = __builtin_amdgcn_wmma_f32_16x16x4_f32(
                        false, af[i], false, bf[j], (short)0, acc[i][j], false, false);
        }
    }

    float bv[4];
#pragma unroll
    for (int j = 0; j < 4; ++j) bv[j] = bout[bj * BN + wn * 64 + j * 16 + lr];
#pragma unroll
    for (int i = 0; i < 2; ++i)
#pragma unroll
        for (int j = 0; j < 4; ++j) {
            const int col = bj * BN + wn * 64 + j * 16 + lr;
#pragma unroll
            for (int r = 0; r < 8; ++r) {
                const int row = bi * BM + wm * 32 + i * 16 + r + 8 * half;
                out[(size_t)row * DIM + col] = acc[i][j][r] + bv[j];
            }
        }
}

// ---------------------------------------------------------------------------
extern "C" void kernel_launch(void* const* d_in, const int* in_sizes, int n_in,
                              void* d_out, int out_size, void* d_ws, size_t ws_size,
                              hipStream_t stream) {
    const float* t_in  = (const float*)d_in[0];
    const float* k_in  = (const float*)d_in[1];
    const float* ln_tg = (const float*)d_in[2];
    const float* ln_tb = (const float*)d_in[3];
    const float* ln_kg = (const float*)d_in[4];
    const float* ln_kb = (const float*)d_in[5];
    const float* W1    = (const float*)d_in[6];
    const float* b1    = (const float*)d_in[7];
    const float* W2    = (const float*)d_in[8];
    const float* b2    = (const float*)d_in[9];
    const float* Wout  = (const float*)d_in[10];
    const float* bout  = (const float*)d_in[11];

    float* ws      = (float*)d_ws;
    float* tn      = ws;
    float* kn      = tn + (size_t)NROWS * DIM;
    float* partial = kn + (size_t)NROWS * DIM;           // [32][N][2]
    float* wts     = partial + (size_t)32 * NROWS * 2;   // [N][2]

    ln_kernel<<<NROWS, 256, 0, stream>>>(t_in, ln_tg, ln_tb, tn);
    ln_kernel<<<NROWS, 256, 0, stream>>>(k_in, ln_kg, ln_kb, kn);

    dim3 grid(DIM / BN, NROWS / BM);  // (32, 64)
    gemm1_kernel<<<grid, 256, 0, stream>>>(tn, kn, W1, b1, W2, partial);
    softmax_kernel<<<NROWS / 256, 256, 0, stream>>>(partial, b2, wts);
    gemm_out_kernel<<<grid, 256, 0, stream>>>(tn, kn, wts, Wout, bout, (float*)d_out);
}